// PostProcessImage_5952824672688
// MI455X (gfx1250) — compile-verified
//
#include <hip/hip_runtime.h>
#include <cstdint>

typedef float        v4f __attribute__((ext_vector_type(4)));
typedef unsigned int v4u __attribute__((ext_vector_type(4)));
typedef int          v4i __attribute__((ext_vector_type(4)));
typedef int          v8i __attribute__((ext_vector_type(8)));

#define B_  2
#define Q_  300
#define C_  80
#define MH_ 128
#define OH_ 512

__device__ __forceinline__ float sigmoidf(float x) {
    return 1.0f / (1.0f + __expf(-x));
}

// ---------------------------------------------------------------------------
// Kernel 1: scores (max over classes of sigmoid*sigmoid), labels (=1), boxes.
// ---------------------------------------------------------------------------
__global__ void ppi_head_kernel(const float* __restrict__ logits,
                                const float* __restrict__ boxes_in,
                                const float* __restrict__ presence,
                                const int*   __restrict__ ts_boxes,
                                float* __restrict__ out) {
    int t = blockIdx.x * blockDim.x + threadIdx.x;
    if (t >= B_ * Q_) return;
    int b = t / Q_;

    const float* lg = logits   + (size_t)t * C_;
    const float* pr = presence + (size_t)b * C_;
    float best = 0.0f;
    #pragma unroll 4
    for (int c = 0; c < C_; ++c) {
        float s = sigmoidf(lg[c]) * sigmoidf(pr[c]);
        best = fmaxf(best, s);
    }
    out[t] = best;                 // scores
    out[B_ * Q_ + t] = 1.0f;       // labels (ones)

    const float* bx = boxes_in + (size_t)t * 4;
    float cx = bx[0], cy = bx[1], w = bx[2], h = bx[3];
    float ih = (float)ts_boxes[b * 2 + 0];
    float iw = (float)ts_boxes[b * 2 + 1];
    float* ob = out + 2 * B_ * Q_ + (size_t)t * 4;
    ob[0] = (cx - 0.5f * w) * iw;
    ob[1] = (cy - 0.5f * h) * ih;
    ob[2] = (cx + 0.5f * w) * iw;
    ob[3] = (cy + 0.5f * h) * ih;
}

// ---------------------------------------------------------------------------
// Kernel 2: 4x bilinear upsample of mask logits + sign threshold.
// grid = 600 masks x 4 bands of 128 output rows. One TDM (Tensor Data Mover)
// descriptor per block DMAs the needed 33-34 input rows into LDS
// (tensor_load_to_lds, tracked by TENSORcnt); compute reads LDS with 16x
// reuse and streams float4 results with nontemporal b128 stores.
// sigmoid(v) > 0.5  <=>  v > 0, so no transcendental needed.
// ---------------------------------------------------------------------------
__global__ void __launch_bounds__(256)
ppi_mask_kernel(const float* __restrict__ masks, float* __restrict__ out) {
    __shared__ float smem[34 * 128];   // 17.4 KB tile: rows [r_lo, r_hi] of mask

    const int bid  = blockIdx.x;
    const int m    = bid >> 2;          // mask index 0..599
    const int band = bid & 3;           // row band 0..3
    const int t    = threadIdx.x;

    // input rows needed by this band: floor coords span [32*band-1, 32*band+32]
    const int r_lo  = (32 * band - 1) < 0 ? 0 : (32 * band - 1);
    const int r_hi  = (32 * band + 32) > 127 ? 127 : (32 * band + 32);
    const int nrows = r_hi - r_lo + 1;  // 33 or 34

    const float*   msrc  = masks + (size_t)m * (MH_ * MH_);
    const uint32_t sbase = (uint32_t)(uintptr_t)(void*)&smem[0];

    // --- Tensor Data Mover: one 2D-tile DMA (nrows x 128 f32) into LDS ------
    if (t < 32) {                       // exactly wave 0 issues the descriptor
        uint64_t ga = (uint64_t)(uintptr_t)(msrc + (size_t)r_lo * MH_);
        // D# group 0: count=1 | lds_addr | global_addr[56:0] | type=2
        v4u g0 = { 1u,
                   sbase,
                   (uint32_t)ga,
                   ((uint32_t)(ga >> 32) & 0x01FFFFFFu) | 0x80000000u };
        // D# group 1: data_size=4B; tensor 128 x (128-r_lo); tile 128 x nrows;
        //             dim0 stride = 128 elements; no pad/iterate/multicast.
        v8i g1 = { (2 << 16),                 // data_size = 2 (4 bytes)
                   (MH_ << 16),               // tensor_dim0 = 128
                   ((MH_ - r_lo) << 16),      // tensor_dim1 = rows available
                   (MH_ << 16),               // tile_dim0 = 128
                   nrows,                     // tile_dim1
                   MH_,                       // tensor_dim0_stride = 128
                   0, 0 };
        v4i z4 = { 0, 0, 0, 0 };
#if defined(__clang_major__) && __clang_major__ >= 23
        v8i z8 = { 0, 0, 0, 0, 0, 0, 0, 0 };
        __builtin_amdgcn_tensor_load_to_lds(g0, g1, z4, z4, z8, 0);
#else
        __builtin_amdgcn_tensor_load_to_lds(g0, g1, z4, z4, 0);
#endif
        __builtin_amdgcn_s_wait_tensorcnt(0);
    }
    __syncthreads();

    // Each thread owns output columns [4i, 4i+3]; input cols needed: i-1, i, i+1.
    const int i  = t & 127;
    const int rp = t >> 7;              // which row of each 2-row group
    const int xm = i > 0   ? i - 1 : 0;
    const int xp = i < 127 ? i + 1 : 127;
    float* obase = out + (size_t)m * (OH_ * OH_) + (size_t)(band * 128) * OH_;

    #pragma unroll 4
    for (int it = 0; it < 64; ++it) {
        int   oy = it * 2 + rp;                          // 0..127 within band
        float cy = (float)(band * 128 + oy) * 0.25f - 0.375f;
        int   y0 = (int)floorf(cy);
        float fy = cy - (float)y0;
        int y0c = y0     < 0 ? 0 : (y0     > 127 ? 127 : y0);
        int y1c = y0 + 1 < 0 ? 0 : (y0 + 1 > 127 ? 127 : y0 + 1);

        const float* r0 = &smem[(y0c - r_lo) * 128];
        const float* r1 = &smem[(y1c - r_lo) * 128];
        float am = r0[xm], a0 = r0[i], ap = r0[xp];
        float bm = r1[xm], b0 = r1[i], bp = r1[xp];

        // vertical lerp once, then the four fixed horizontal weights
        float mm = am + fy * (bm - am);
        float m0 = a0 + fy * (b0 - a0);
        float mp = ap + fy * (bp - ap);

        v4f v;
        v.x = (mm + 0.625f * (m0 - mm)) > 0.0f ? 1.0f : 0.0f;
        v.y = (mm + 0.875f * (m0 - mm)) > 0.0f ? 1.0f : 0.0f;
        v.z = (m0 + 0.125f * (mp - m0)) > 0.0f ? 1.0f : 0.0f;
        v.w = (m0 + 0.375f * (mp - m0)) > 0.0f ? 1.0f : 0.0f;
        __builtin_nontemporal_store(v, (v4f*)(obase + (size_t)oy * OH_ + i * 4));
    }
}

// ---------------------------------------------------------------------------
extern "C" void kernel_launch(void* const* d_in, const int* in_sizes, int n_in,
                              void* d_out, int out_size, void* d_ws, size_t ws_size,
                              hipStream_t stream) {
    const float* pred_logits = (const float*)d_in[0];   // [2,300,80]
    const float* pred_boxes  = (const float*)d_in[1];   // [2,300,4]
    const float* pred_masks  = (const float*)d_in[2];   // [2,300,128,128]
    const float* presence    = (const float*)d_in[3];   // [2,80]
    const int*   ts_boxes    = (const int*)d_in[4];     // [2,2]
    // d_in[5] (target_sizes_masks) is static 512x512 in the reference path.

    float* out = (float*)d_out;
    // Layout: scores[600] | labels[600] | boxes[2400] | masks[600*512*512]

    ppi_head_kernel<<<dim3(3), dim3(256), 0, stream>>>(
        pred_logits, pred_boxes, presence, ts_boxes, out);

    ppi_mask_kernel<<<dim3(600 * 4), dim3(256), 0, stream>>>(
        pred_masks, out + 3600);
}